// Cal_Div_Loss_11802570130034
// MI455X (gfx1250) — compile-verified
//
#include <hip/hip_runtime.h>

// ---------------------------------------------------------------------------
// Cal_Div_Loss on MI455X (gfx1250, wave32)
//
// conv(x)-conv(y) == conv(x-y)  ->  work on d = x - y only.
//   pass1: stream x,y once (128MB @ 23.3TB/s -> ~5.5us floor), b128 NT loads,
//          LDS-tile 66x66, emit level-0 partials (|d|, d) and the level-1
//          255x255 map (+ |L1| partials). NT hints keep d1 L2-resident.
//   pass2..4: pyramid 255->127->63->31 entirely out of L2/workspace (tiny)
//   pass5: deterministic fixed-order fold of partials + stateful 'fuhao' signs
// Block reductions: V_WMMA_F32_16X16X4_F32 ones-matrix trick (layout-robust).
// No float atomics anywhere -> bit-deterministic across graph replays.
// ---------------------------------------------------------------------------

typedef __attribute__((ext_vector_type(2))) float v2f;
typedef __attribute__((ext_vector_type(4))) float v4f;
typedef __attribute__((ext_vector_type(8))) float v8f;

#define IMG_H   512
#define IMG_W   512
#define NPIX    (IMG_H * IMG_W)
#define DIM1    255
#define DIM2    127
#define DIM3    63
#define DIM4    31
#define NB      64

// workspace layout (float offsets)
#define WS_D1   ((size_t)0)                    // 64*255*255 = 4161600
#define WS_D2   ((size_t)4161600)              // 64*127*127 = 1032256
#define WS_D3   ((size_t)5193856)              // 64*63*63   = 254016
#define WS_P0   ((size_t)5447872)              // 64*64 level-0 |d| partials
#define WS_PD   ((size_t)5451968)              // 64*64 level-0 d partials
#define WS_P1   ((size_t)5456064)              // 64*64 level-1 |.| partials
#define WS_P2   ((size_t)5460160)              // 64*8
#define WS_P3   ((size_t)5460672)              // 64*2
#define WS_P4   ((size_t)5460800)              // 64*1

// Sum 256 floats (LDS array) with one wave via V_WMMA_F32_16X16X4_F32.
// A = ones(16x4), B = data chunk (4x16), accumulate C over 4 chunks:
//   D[m][n] = sum_k B[k][n]  ->  sum_n D[0][n] = total of all 64 values,
// independent of the exact B VGPR striping (any bijection works).
// Must be called by a full wave (EXEC all ones). Lane 0 returns the sum.
__device__ __forceinline__ float wmma_reduce256(const float* __restrict__ s) {
  const int lane = (int)(threadIdx.x & 31u);
  const int half = lane >> 4;       // 0: lanes 0-15, 1: lanes 16-31
  const int n    = lane & 15;
  v2f a; a.x = 1.0f; a.y = 1.0f;    // all-ones A matrix (16x4)
  v8f c = {};
#pragma unroll
  for (int chunk = 0; chunk < 4; ++chunk) {
    v2f bm;
    bm.x = s[chunk * 64 + half * 32 + n];
    bm.y = s[chunk * 64 + half * 32 + 16 + n];
    c = __builtin_amdgcn_wmma_f32_16x16x4_f32(
        /*neg_a=*/false, a, /*neg_b=*/false, bm,
        /*c_mod=*/(short)0, c, /*reuse_a=*/false, /*reuse_b=*/false);
  }
  // c[0] on lanes 0..15 holds D[0][n]; sum the 16-lane group.
  float v = c[0];
  v += __shfl_xor(v, 1, 32);
  v += __shfl_xor(v, 2, 32);
  v += __shfl_xor(v, 4, 32);
  v += __shfl_xor(v, 8, 32);
  return v;   // lane 0: total
}

// ---------------------------------------------------------------------------
// pass1: grid (8,8,64), block 256.  Each WG owns a 64x64 input region (+2 halo)
// and the matching 32x32 tile of the level-1 map.
// ---------------------------------------------------------------------------
__global__ __launch_bounds__(256) void pass1_kernel(
    const float* __restrict__ x, const float* __restrict__ y,
    float* __restrict__ d1, float* __restrict__ p0,
    float* __restrict__ pd, float* __restrict__ p1) {
  __shared__ float sd[66][67];     // d = x - y tile (+1 col pad vs bank conflicts)
  __shared__ float red[3][256];

  const int wx = blockIdx.x, wy = blockIdx.y, b = blockIdx.z;
  const int tid = (int)threadIdx.x;
  const int r0 = wy * 64, c0 = wx * 64;
  const size_t base = (size_t)b * NPIX;

  // ---- tile body: 66 rows x 64 cols as float4 (16B-aligned), NT streaming ----
  for (int idx = tid; idx < 66 * 16; idx += 256) {
    const int rr = idx >> 4, qq = idx & 15;
    const int r = r0 + rr, c = c0 + (qq << 2);
    v4f vx = {0.0f, 0.0f, 0.0f, 0.0f};
    v4f vy = {0.0f, 0.0f, 0.0f, 0.0f};
    if (r < IMG_H) {                       // c+3 <= 511 always (c0 <= 448)
      const size_t g = base + (size_t)r * IMG_W + (size_t)c;
      vx = __builtin_nontemporal_load((const v4f*)(x + g));
      vy = __builtin_nontemporal_load((const v4f*)(y + g));
    }
    const v4f vd = vx - vy;
    sd[rr][c - c0 + 0] = vd.x;
    sd[rr][c - c0 + 1] = vd.y;
    sd[rr][c - c0 + 2] = vd.z;
    sd[rr][c - c0 + 3] = vd.w;
  }
  // ---- halo: 2 extra columns (cols 64,65 of the tile) ----
  for (int idx = tid; idx < 66 * 2; idx += 256) {
    const int rr = idx >> 1, h = idx & 1;
    const int r = r0 + rr, c = c0 + 64 + h;
    float v = 0.0f;
    if (r < IMG_H && c < IMG_W) {
      const size_t g = base + (size_t)r * IMG_W + (size_t)c;
      v = __builtin_nontemporal_load(x + g) - __builtin_nontemporal_load(y + g);
    }
    sd[rr][64 + h] = v;
  }
  __syncthreads();

  // level-0 partials over the owned (non-halo) 64x64 region
  float a0 = 0.0f, ad = 0.0f;
  for (int idx = tid; idx < 64 * 64; idx += 256) {
    const int rr = idx >> 6, cc = idx & 63;
    const float v = sd[rr][cc];
    a0 += fabsf(v);
    ad += v;
  }

  // level-1: 3x3 sum-pool stride 2 -> up to 32x32 outputs
  float a1 = 0.0f;
  for (int idx = tid; idx < 32 * 32; idx += 256) {
    const int i = idx >> 5, j = idx & 31;
    const int oy = wy * 32 + i, ox = wx * 32 + j;
    if (oy < DIM1 && ox < DIM1) {
      const int li = 2 * i, lj = 2 * j;
      const float s =
          sd[li    ][lj] + sd[li    ][lj + 1] + sd[li    ][lj + 2] +
          sd[li + 1][lj] + sd[li + 1][lj + 1] + sd[li + 1][lj + 2] +
          sd[li + 2][lj] + sd[li + 2][lj + 1] + sd[li + 2][lj + 2];
      d1[(size_t)b * (DIM1 * DIM1) + (size_t)oy * DIM1 + ox] = s;
      a1 += fabsf(s);
    }
  }

  red[0][tid] = a0;
  red[1][tid] = ad;
  red[2][tid] = a1;
  __syncthreads();

  const int wid = tid >> 5;
  if (wid < 3) {                      // waves 0,1,2 each reduce one array (WMMA)
    const float v = wmma_reduce256(&red[wid][0]);
    if ((tid & 31) == 0) {
      const int part = (b * 8 + wy) * 8 + wx;       // unique slot per WG
      float* dst = (wid == 0) ? p0 : (wid == 1) ? pd : p1;
      dst[part] = v;
    }
  }
}

// ---------------------------------------------------------------------------
// pyramid pass: IN x IN -> OUT x OUT 3x3 sum-pool stride 2, |out| partial per WG
// grid (NWG, 64), block 256.  Input stays L2-resident (<=16.6MB total).
// ---------------------------------------------------------------------------
template <int IN, int OUT, int ROWS, bool STORE>
__global__ __launch_bounds__(256) void convpass_kernel(
    const float* __restrict__ din, float* __restrict__ dout,
    float* __restrict__ part, int nwg) {
  __shared__ float red[256];
  const int wy = blockIdx.x, b = blockIdx.y;
  const int tid = (int)threadIdx.x;
  const float* in = din + (size_t)b * (IN * IN);

  float a = 0.0f;
  const int row0 = wy * ROWS;
  const int nrows = (OUT - row0 < ROWS) ? (OUT - row0) : ROWS;
  const int total = nrows * OUT;
  for (int idx = tid; idx < total; idx += 256) {
    const int i = idx / OUT, j = idx - i * OUT;
    const int oy = row0 + i;
    const float* r = in + (size_t)(2 * oy) * IN + (size_t)(2 * j);
    const float s = r[0]          + r[1]          + r[2] +
                    r[IN]         + r[IN + 1]     + r[IN + 2] +
                    r[2 * IN]     + r[2 * IN + 1] + r[2 * IN + 2];
    if (STORE)
      dout[(size_t)b * (OUT * OUT) + (size_t)oy * OUT + j] = s;
    a += fabsf(s);
  }

  red[tid] = a;
  __syncthreads();
  if (tid < 32) {
    const float v = wmma_reduce256(red);
    if (tid == 0) part[b * nwg + wy] = v;
  }
}

// ---------------------------------------------------------------------------
// final combine: fixed-order partial fold + faithful stateful 'fuhao' logic
// ---------------------------------------------------------------------------
__global__ void combine_kernel(
    const float* __restrict__ alpha,
    const float* __restrict__ p0, const float* __restrict__ pd,
    const float* __restrict__ p1, const float* __restrict__ p2,
    const float* __restrict__ p3, const float* __restrict__ p4,
    float* __restrict__ out) {
  if (threadIdx.x != 0 || blockIdx.x != 0) return;
  double total = 0.0;
  bool triggered = false;                 // 'fuhao' state, never resets
  for (int b = 0; b < NB; ++b) {
    float s0 = 0.0f, sdv = 0.0f, s1 = 0.0f;
    for (int w = 0; w < 64; ++w) {
      s0  += p0[b * 64 + w];
      sdv += pd[b * 64 + w];
      s1  += p1[b * 64 + w];
    }
    float s2 = 0.0f;
    for (int w = 0; w < 8; ++w) s2 += p2[b * 8 + w];
    const float s3 = p3[b * 2 + 0] + p3[b * 2 + 1];
    const float s4 = p4[b];

    const float l1v[5] = {
        s0 / (float)NPIX,           s1 / (float)(DIM1 * DIM1),
        s2 / (float)(DIM2 * DIM2),  s3 / (float)(DIM3 * DIM3),
        s4 / (float)(DIM4 * DIM4)};

    const int k = (int)(alpha[b] * 6.0f);   // trunc, alpha >= 0
    for (int i = 0; i < 5; ++i) {
      const float sgn = (triggered || i >= k) ? 1.0f : -1.0f;
      total += (double)(sgn * l1v[i]);
    }
    total += (double)(fabsf(sdv) * 5.0f);   // last column
    if (k <= 4) triggered = true;           // trig for subsequent rows only
  }
  out[0] = (float)(total / (double)(NB * 6));
}

// ---------------------------------------------------------------------------
extern "C" void kernel_launch(void* const* d_in, const int* in_sizes, int n_in,
                              void* d_out, int out_size, void* d_ws, size_t ws_size,
                              hipStream_t stream) {
  const float* x     = (const float*)d_in[0];
  const float* y     = (const float*)d_in[1];
  const float* alpha = (const float*)d_in[2];
  float* ws  = (float*)d_ws;
  float* out = (float*)d_out;

  float* d1 = ws + WS_D1;
  float* d2 = ws + WS_D2;
  float* d3 = ws + WS_D3;
  float* p0 = ws + WS_P0;
  float* pd = ws + WS_PD;
  float* p1 = ws + WS_P1;
  float* p2 = ws + WS_P2;
  float* p3 = ws + WS_P3;
  float* p4 = ws + WS_P4;

  pass1_kernel<<<dim3(8, 8, NB), 256, 0, stream>>>(x, y, d1, p0, pd, p1);
  convpass_kernel<DIM1, DIM2, 16, true ><<<dim3(8, NB), 256, 0, stream>>>(d1, d2, p2, 8);
  convpass_kernel<DIM2, DIM3, 32, true ><<<dim3(2, NB), 256, 0, stream>>>(d2, d3, p3, 2);
  convpass_kernel<DIM3, DIM4, 31, false><<<dim3(1, NB), 256, 0, stream>>>(d3, nullptr, p4, 1);
  combine_kernel<<<1, 32, 0, stream>>>(alpha, p0, pd, p1, p2, p3, p4, out);
}